// MixedRON_20770461844168
// MI455X (gfx1250) — compile-verified
//
#include <hip/hip_runtime.h>
#include <math.h>

// Problem constants (from reference)
#define BATCH   32
#define TSTEPS  512
#define NI      128
#define NH      1024
#define KTOT    (NI + NH)      // 1152
#define DT_     0.042f
#define UDECAY  (1.0f - DT_)   // 1 - R*C*dt
#define HARM    512.0f         // N_HID * P
#define SPKC    512.0f         // N_HID - HARMONIC
#define THR     0.5f

// Kernel geometry
#define NWG     32
#define BLOCK   128            // 4 waves of 32
#define KCHUNK  384            // 1152 = 3 * 384
#define KT_CH   (KCHUNK / 4)   // 96 K=4 tiles per chunk
#define ASTRIDE 404            // floats: %4==0 (float4), %64==20 (bank spread)

typedef float v2f __attribute__((ext_vector_type(2)));
typedef float v8f __attribute__((ext_vector_type(8)));

__device__ __forceinline__ v8f wmma_k4(v2f a, v2f b, v8f c) {
    return __builtin_amdgcn_wmma_f32_16x16x4_f32(false, a, false, b,
                                                 (short)0, c, false, false);
}

__launch_bounds__(BLOCK, 1)
__global__ void mixed_ron_persistent(const float* __restrict__ x,
                                     const float* __restrict__ x2h,
                                     const float* __restrict__ h2h,
                                     const float* __restrict__ gamma_,
                                     const float* __restrict__ eps_,
                                     const float* __restrict__ bias_,
                                     float* __restrict__ out,
                                     unsigned* __restrict__ bar_cnt)
{
    __shared__ float Alds[BATCH * ASTRIDE];

    const int tid    = threadIdx.x;
    const int lane   = tid & 31;
    const int wave   = tid >> 5;                    // 0..3
    const int m_tile = wave >> 1;                   // batch tile (16 rows)
    const int n_slab = blockIdx.x * 2 + (wave & 1); // 0..63
    const int ncol   = n_slab * 16 + (lane & 15);
    const int khalf  = (lane >> 4) << 1;            // 0 | 2 (K within tile)

    // Staging decomposition: 4 threads per batch row
    const int srow = tid >> 2;                      // 0..31
    const int ssub = tid & 3;                       // 0..3

    const float g_n = gamma_[ncol];
    const float e_n = eps_[ncol];
    const float b_n = bias_[ncol];

    float hy_s[8], hz_s[8], u_s[8];
    #pragma unroll
    for (int j = 0; j < 8; ++j) { hy_s[j] = 0.f; hz_s[j] = 0.f; u_s[j] = 0.f; }

    const size_t SEC = (size_t)BATCH * TSTEPS * NH;
    float* out_hy = out;
    float* out_hz = out + SEC;
    float* out_u  = out + 2 * SEC;
    float* out_sp = out + 3 * SEC;

    const float* aRow    = &Alds[(m_tile * 16 + (lane & 15)) * ASTRIDE];
    float*       AldsRow = &Alds[srow * ASTRIDE];
    const float4 zero4   = make_float4(0.f, 0.f, 0.f, 0.f);

    for (int t = 0; t < TSTEPS; ++t) {
        const float* xrow  = x + (size_t)(srow * TSTEPS + t) * NI;
        const float* hyrow = out_hy + (size_t)(srow * TSTEPS + (t > 0 ? t - 1 : 0)) * NH;

        v8f acc = {0.f, 0.f, 0.f, 0.f, 0.f, 0.f, 0.f, 0.f};

        for (int kc = 0; kc < KTOT; kc += KCHUNK) {
            // -------- stage A chunk = [x_t | hy_{t-1}] rows into LDS --------
            if (kc == 0) {
                #pragma unroll
                for (int q = ssub; q < NI / 4; q += 4)          // x part
                    *(float4*)(AldsRow + q * 4) = *(const float4*)(xrow + q * 4);
                if (t == 0) {
                    #pragma unroll
                    for (int q = ssub; q < (KCHUNK - NI) / 4; q += 4)
                        *(float4*)(AldsRow + NI + q * 4) = zero4;
                } else {
                    #pragma unroll
                    for (int q = ssub; q < (KCHUNK - NI) / 4; q += 4)
                        *(float4*)(AldsRow + NI + q * 4) =
                            *(const float4*)(hyrow + q * 4);
                }
            } else {
                if (t == 0) {
                    #pragma unroll
                    for (int q = ssub; q < KT_CH; q += 4)
                        *(float4*)(AldsRow + q * 4) = zero4;
                } else {
                    const float* src = hyrow + (kc - NI);
                    #pragma unroll
                    for (int q = ssub; q < KT_CH; q += 4)
                        *(float4*)(AldsRow + q * 4) =
                            *(const float4*)(src + q * 4);
                }
            }
            if (kc == 0 && t + 1 < TSTEPS)
                __builtin_prefetch(x + ((size_t)((tid & 31) * TSTEPS + (t + 1)) * NI), 0, 3);
            __syncthreads();

            // -------- WMMA over this chunk; branch-free inner loops --------
            if (kc == 0) {
                // K in [0, 128): B from x2h  (32 tiles)
                int boff = khalf * NH + ncol;
                #pragma unroll 8
                for (int kt = 0; kt < NI / 4; ++kt) {
                    v2f a = *(const v2f*)(aRow + kt * 4 + khalf);
                    v2f b; b.x = x2h[boff]; b.y = x2h[boff + NH];
                    boff += 4 * NH;
                    acc = wmma_k4(a, b, acc);
                }
                // K in [128, 384): B from h2h rows [0, 256)  (64 tiles)
                int hoff = khalf * NH + ncol;
                #pragma unroll 8
                for (int kt = NI / 4; kt < KT_CH; ++kt) {
                    v2f a = *(const v2f*)(aRow + kt * 4 + khalf);
                    v2f b; b.x = h2h[hoff]; b.y = h2h[hoff + NH];
                    hoff += 4 * NH;
                    acc = wmma_k4(a, b, acc);
                }
            } else {
                // B from h2h rows [kc-128, kc+256)  (96 tiles)
                int hoff = (kc - NI + khalf) * NH + ncol;
                #pragma unroll 8
                for (int kt = 0; kt < KT_CH; ++kt) {
                    v2f a = *(const v2f*)(aRow + kt * 4 + khalf);
                    v2f b; b.x = h2h[hoff]; b.y = h2h[hoff + NH];
                    hoff += 4 * NH;
                    acc = wmma_k4(a, b, acc);
                }
            }
            __syncthreads();   // all waves done with LDS before restage
        }

        // -------- elementwise oscillator update + stream 4 outputs --------
        #pragma unroll
        for (int j = 0; j < 8; ++j) {
            int brow = m_tile * 16 + ((lane < 16) ? j : j + 8);
            float u  = u_s[j];
            float hy = hy_s[j];
            float hz = hz_s[j];
            float spike = (u > THR) ? 1.f : 0.f;
            u = (u > THR) ? 0.f : u;
            u *= UDECAY;
            float drive = tanhf(acc[j] + b_n);
            hz = hz + DT_ * (drive - g_n * hy - (e_n * (hz * HARM) + u * SPKC));
            hy = hy + DT_ * hz;
            size_t o = (size_t)(brow * TSTEPS + t) * NH + ncol;
            out_hy[o] = hy;
            out_hz[o] = hz;
            out_u[o]  = u;
            out_sp[o] = spike;
            hy_s[j] = hy; hz_s[j] = hz; u_s[j] = u;
        }

        // -------- device-wide barrier: publish hy_t before step t+1 --------
        __threadfence();                       // release hy stores
        __syncthreads();
        if (tid == 0) {
            __hip_atomic_fetch_add(bar_cnt, 1u, __ATOMIC_RELEASE,
                                   __HIP_MEMORY_SCOPE_AGENT);
            unsigned target = (unsigned)(NWG * (t + 1));
            while (__hip_atomic_load(bar_cnt, __ATOMIC_ACQUIRE,
                                     __HIP_MEMORY_SCOPE_AGENT) < target) {
                __builtin_amdgcn_s_sleep(2);
            }
        }
        __syncthreads();
        __threadfence();                       // acquire: invalidate L0
    }
}

extern "C" void kernel_launch(void* const* d_in, const int* in_sizes, int n_in,
                              void* d_out, int out_size, void* d_ws, size_t ws_size,
                              hipStream_t stream) {
    (void)in_sizes; (void)n_in; (void)out_size;
    const float* x    = (const float*)d_in[0];
    const float* x2h  = (const float*)d_in[1];
    const float* h2h  = (const float*)d_in[2];
    const float* gam  = (const float*)d_in[3];
    const float* eps  = (const float*)d_in[4];
    const float* bias = (const float*)d_in[5];

    if (ws_size >= 256) hipMemsetAsync(d_ws, 0, 256, stream);

    hipLaunchKernelGGL(mixed_ron_persistent, dim3(NWG), dim3(BLOCK), 0, stream,
                       x, x2h, h2h, gam, eps, bias,
                       (float*)d_out, (unsigned*)d_ws);
}